// GraphHardCounter_40020505264198
// MI455X (gfx1250) — compile-verified
//
#include <hip/hip_runtime.h>

#define THREADS 256
#define EPT 16        // edges per thread
#define WINDOW 64     // graphs covered by the block-local LDS accumulator

__global__ void zero_f32(float* __restrict__ p, int n) {
    int i = blockIdx.x * blockDim.x + threadIdx.x;
    if (i < n) p[i] = 0.0f;
}

__launch_bounds__(THREADS)
__global__ void graph_edge_score_sum(
    const int* __restrict__ node_type,
    const int* __restrict__ edge_type,
    const int* __restrict__ edge_index,   // [2, E] row-major
    const int* __restrict__ edge_batch,   // sorted
    const float* __restrict__ w,          // [n_tab] scorer table (450)
    float* __restrict__ out,              // [G], pre-zeroed
    int E, int G, int n_tab)
{
    __shared__ float tbl[512];
    __shared__ float acc[WINDOW];
    __shared__ int   s_first;

    const int tid = threadIdx.x;
    if (tid < WINDOW) acc[tid] = 0.0f;

    // --- Stage the scorer table into LDS with gfx1250 async global->LDS copies.
    for (int i = tid; i < n_tab; i += THREADS) {
        unsigned lds_off = (unsigned)(unsigned long long)(&tbl[i]); // flat->LDS: low 32 bits
        const float* gp = w + i;
        asm volatile("global_load_async_to_lds_b32 %0, %1, off"
                     :: "v"(lds_off), "v"(gp) : "memory");
    }
    asm volatile("s_wait_asynccnt 0" ::: "memory");

    const long long base = (long long)blockIdx.x * (THREADS * EPT);
    if (tid == 0) s_first = edge_batch[base];
    __syncthreads();   // table staged, window zeroed, s_first visible

    const int first_g = s_first;
    const long long e0 = base + (long long)tid * EPT;

    float run = 0.0f;
    int   rg  = -1;

    auto flush = [&](int g, float v) {
        if (g < 0) return;
        int off = g - first_g;
        if (0 <= off && off < WINDOW) {
            atomicAdd(&acc[off], v);          // ds_add_f32, block-local
        } else {
            atomicAdd(&out[g], v);            // rare spill (many tiny graphs)
        }
    };

    if ((E % 4 == 0) && (e0 + EPT <= (long long)E)) {
        // Fast path: b128 coalesced stream loads, 16 edges per thread.
        const int4* et4 = (const int4*)(edge_type)      + (e0 >> 2);
        const int4* sr4 = (const int4*)(edge_index)     + (e0 >> 2);
        const int4* dd4 = (const int4*)(edge_index + E) + (e0 >> 2);
        const int4* eb4 = (const int4*)(edge_batch)     + (e0 >> 2);
        #pragma unroll
        for (int q = 0; q < EPT / 4; ++q) {
            int4 et = et4[q];
            int4 sr = sr4[q];
            int4 dd = dd4[q];
            int4 eb = eb4[q];
            int enc0 = et.x * 9 + node_type[sr.x] * 3 + node_type[dd.x];
            int enc1 = et.y * 9 + node_type[sr.y] * 3 + node_type[dd.y];
            int enc2 = et.z * 9 + node_type[sr.z] * 3 + node_type[dd.z];
            int enc3 = et.w * 9 + node_type[sr.w] * 3 + node_type[dd.w];
            int   bb[4] = {eb.x, eb.y, eb.z, eb.w};
            float vv[4] = {tbl[enc0], tbl[enc1], tbl[enc2], tbl[enc3]};
            #pragma unroll
            for (int j = 0; j < 4; ++j) {
                if (bb[j] == rg) {
                    run += vv[j];
                } else {
                    flush(rg, run);
                    rg = bb[j];
                    run = vv[j];
                }
            }
        }
    } else {
        // Generic tail path.
        long long eEnd = e0 + EPT;
        if (eEnd > (long long)E) eEnd = E;
        for (long long e = e0; e < eEnd; ++e) {
            int et = edge_type[e];
            int s  = edge_index[e];
            int d  = edge_index[(long long)E + e];
            int b  = edge_batch[e];
            float v = tbl[et * 9 + node_type[s] * 3 + node_type[d]];
            if (b == rg) {
                run += v;
            } else {
                flush(rg, run);
                rg = b;
                run = v;
            }
        }
    }
    flush(rg, run);
    __syncthreads();

    // Write the block-local window back (<= 64 global atomics per block).
    for (int i = tid; i < WINDOW; i += THREADS) {
        float v = acc[i];
        int g = first_g + i;
        if (g < G && v != 0.0f) atomicAdd(&out[g], v);
    }
}

extern "C" void kernel_launch(void* const* d_in, const int* in_sizes, int n_in,
                              void* d_out, int out_size, void* d_ws, size_t ws_size,
                              hipStream_t stream) {
    const int*   node_type  = (const int*)d_in[0];
    const int*   edge_type  = (const int*)d_in[1];
    const int*   edge_index = (const int*)d_in[2];
    const int*   edge_batch = (const int*)d_in[3];
    const float* w          = (const float*)d_in[4];
    float*       out        = (float*)d_out;

    const int E     = in_sizes[1];
    const int G     = out_size;       // [G,1] output, flat G floats
    const int n_tab = in_sizes[4];

    zero_f32<<<(G + 255) / 256, 256, 0, stream>>>(out, G);

    const int per_block = THREADS * EPT;
    const int blocks = (E + per_block - 1) / per_block;
    graph_edge_score_sum<<<blocks, THREADS, 0, stream>>>(
        node_type, edge_type, edge_index, edge_batch, w, out, E, G, n_tab);
}